// SlowNorm_75977971466795
// MI455X (gfx1250) — compile-verified
//
#include <hip/hip_runtime.h>
#include <stdint.h>

// irreps: (256,1)+(256,3)+(128,5)+(64,7) -> DIM=2112 in, OUTW=704 out per row
#define DIM     2112
#define OUTW    704
#define ROWS    4                    // rows staged per block
#define THREADS 256                  // 8 wave32s
#define TILE_FLOATS (ROWS * DIM)     // 8448 floats = 33792 B LDS
#define CHUNKS  (TILE_FLOATS / 4)    // 2112 x 16B async copies per tile

// One async b128 copy: 16 B/lane from global (sbase plus voff) to LDS (ldsa), both plus OFF.
// The immediate offset is applied to BOTH the LDS destination and the global
// source address (CDNA5 ISA), so one (voff, ldsa) pair serves
// the whole tile.
#define STAGE(OFF)                                                          \
    asm volatile("global_load_async_to_lds_b128 %0, %1, %2 offset:" #OFF    \
                 :: "v"(ldsa), "v"(voff), "s"(sbase) : "memory")

__global__ __launch_bounds__(THREADS)
void SlowNorm_kernel(const float* __restrict__ in, float* __restrict__ out, int batch)
{
    __shared__ float tile[TILE_FLOATS];

    const int tid = threadIdx.x;
    const long long row_base = (long long)blockIdx.x * ROWS;
    if (row_base >= batch) return;

    // ---- Stage: global -> LDS via CDNA5 async copy engine (ASYNCcnt) ----
    const unsigned long long sbase =
        (unsigned long long)(uintptr_t)(in + row_base * DIM);
    // Low 32 bits of a generic pointer into LDS = byte offset in the LDS alloc.
    const unsigned voff = (unsigned)tid * 16u;
    const unsigned ldsa = (unsigned)(uintptr_t)(&tile[0]) + voff;

    if (row_base + ROWS <= batch) {
        // Full tile: 2112 chunks = 8*256 + 64. 256 lanes advance 4096 B/issue.
        STAGE(0);     STAGE(4096);  STAGE(8192);  STAGE(12288);
        STAGE(16384); STAGE(20480); STAGE(24576); STAGE(28672);
        if (tid < (CHUNKS - 8 * THREADS)) { STAGE(32768); }
    } else {
        // Generic tail tile (not hit for batch=65536, kept for safety).
        const int chunks_valid = (int)((batch - row_base) * DIM / 4);
        for (int c = tid; c < chunks_valid; c += THREADS) {
            unsigned vo = (unsigned)c * 16u;
            unsigned la = (unsigned)(uintptr_t)(&tile[0]) + vo;
            asm volatile("global_load_async_to_lds_b128 %0, %1, %2 offset:0"
                         :: "v"(la), "v"(vo), "s"(sbase) : "memory");
        }
    }
#if __has_builtin(__builtin_amdgcn_s_wait_asynccnt)
    __builtin_amdgcn_s_wait_asynccnt(0);
#else
    asm volatile("s_wait_asynccnt 0" ::: "memory");
#endif
    __syncthreads();

    // ---- Compute: per-segment loops with compile-time d ----
    // LDS strides 1/3/5/7 dwords are coprime with 64 banks: conflict-free.

    // Segment l=0: d=1, 4 rows x 256 cols, col j = tid
    #pragma unroll
    for (int i = 0; i < ROWS; ++i) {
        float x = tile[i * DIM + tid];
        if (row_base + i < batch)
            out[(row_base + i) * OUTW + tid] = __builtin_sqrtf(x * x);
    }

    // Segment l=1: d=3, 4 rows x 256 cols, col j = 256 + tid
    #pragma unroll
    for (int i = 0; i < ROWS; ++i) {
        const float* p = &tile[i * DIM + 256 + tid * 3];
        float s = p[0] * p[0];
        s = __builtin_fmaf(p[1], p[1], s);
        s = __builtin_fmaf(p[2], p[2], s);
        if (row_base + i < batch)
            out[(row_base + i) * OUTW + 256 + tid] = __builtin_sqrtf(s);
    }

    // Segment l=2: d=5, 4 rows x 128 cols -> 2 iterations of 256 threads
    #pragma unroll
    for (int i = 0; i < 2; ++i) {
        int r = 2 * i + (tid >> 7);
        int m = tid & 127;
        const float* p = &tile[r * DIM + 1024 + m * 5];
        float s = p[0] * p[0];
        s = __builtin_fmaf(p[1], p[1], s);
        s = __builtin_fmaf(p[2], p[2], s);
        s = __builtin_fmaf(p[3], p[3], s);
        s = __builtin_fmaf(p[4], p[4], s);
        if (row_base + r < batch)
            out[(row_base + r) * OUTW + 512 + m] = __builtin_sqrtf(s);
    }

    // Segment l=3: d=7, 4 rows x 64 cols -> 1 iteration of 256 threads
    {
        int r = tid >> 6;
        int m = tid & 63;
        const float* p = &tile[r * DIM + 1664 + m * 7];
        float s = p[0] * p[0];
        s = __builtin_fmaf(p[1], p[1], s);
        s = __builtin_fmaf(p[2], p[2], s);
        s = __builtin_fmaf(p[3], p[3], s);
        s = __builtin_fmaf(p[4], p[4], s);
        s = __builtin_fmaf(p[5], p[5], s);
        s = __builtin_fmaf(p[6], p[6], s);
        if (row_base + r < batch)
            out[(row_base + r) * OUTW + 640 + m] = __builtin_sqrtf(s);
    }
}

extern "C" void kernel_launch(void* const* d_in, const int* in_sizes, int n_in,
                              void* d_out, int out_size, void* d_ws, size_t ws_size,
                              hipStream_t stream)
{
    const float* features = (const float*)d_in[0];
    float* out = (float*)d_out;

    const int batch = in_sizes[0] / DIM;              // 65536
    const int grid  = (batch + ROWS - 1) / ROWS;      // 16384 blocks

    SlowNorm_kernel<<<grid, THREADS, 0, stream>>>(features, out, batch);
}